// Invert_4157528342908
// MI455X (gfx1250) — compile-verified
//
#include <hip/hip_runtime.h>

// Reference factorization:
//   path_b = x[b] * P  where P is fixed (from W prefix sums)
//   => out[b, j] = x[b]^level(j) * sig[j],  sig = signature of P (126 values)
//
// Kernel 1: compute sig once (chunked parallel Chen scan, associativity).
//           Emit M[8][128]: M[k][j] = sig[j] if level(j)==k else 0.
// Kernel 2: out(4096x126) = V(4096x8) @ M(8x128), V[b,k]=x[b]^k,
//           via two chained V_WMMA_F32_16X16X4_F32 per 16x16 tile.

#define NINC     511     // number of path increments
#define SIG_DIM  126     // 2+4+8+16+32+64
#define CHUNKS   16
#define CHUNK_LEN 32     // 16*32 = 512 >= 511

typedef __attribute__((ext_vector_type(2))) float v2f;
typedef __attribute__((ext_vector_type(8))) float v8f;

// Element t (0..125) -> level k (1..6), local index j (0..2^k-1).
// Level k occupies [2^k - 2, 2^{k+1} - 2).
__device__ __forceinline__ void decode_elem(int t, int& k, int& j) {
    int v = t + 2;
    k = 31 - __clz(v);
    j = v - (1 << k);
}

// One Chen step with a single increment dx = (dx0, dx1):
//   new_k[j] = A_k[j] + E_k[j] + sum_{i=1}^{k-1} A_i[prefix_i(j)] * E_{k-i}[suffix(j)]
// E levels built incrementally from the suffix digits: E_len = E_{len-1}*dx_d/len.
__device__ __forceinline__
float chen_step_inc(const float* A, int t, int k, int j,
                    float dx0, float dx1, bool first) {
    float val = first ? 0.0f : A[t];
    float e = 1.0f;
    float len = 0.0f;
    for (int i = k - 1; i >= 1; --i) {
        int d = (j >> (k - 1 - i)) & 1;      // digit i+1 (suffix side)
        len += 1.0f;
        e = e * (d ? dx1 : dx0) / len;       // e = E_{k-i}[suffix digits]
        if (!first) {
            int pref = (1 << i) - 2 + (j >> (k - i));   // global idx of A_i prefix
            val += A[pref] * e;
        }
    }
    int d1 = (j >> (k - 1)) & 1;             // digit 1
    len += 1.0f;
    e = e * (d1 ? dx1 : dx0) / len;          // e = E_k[j]
    return val + e;
}

// Chen combine of two full signatures: C_k = A_k + B_k + sum A_i (x) B_{k-i}.
__device__ __forceinline__
float chen_combine_sig(const float* A, const float* B, int t, int k, int j) {
    float val = A[t] + B[t];
    for (int i = 1; i < k; ++i) {
        int pref = (1 << i) - 2 + (j >> (k - i));
        int sufl = k - i;
        int suf  = (1 << sufl) - 2 + (j & ((1 << sufl) - 1));
        val += A[pref] * B[suf];
    }
    return val;
}

__global__ __launch_bounds__(1024)
void sig_shared_kernel(const float* __restrict__ W, float* __restrict__ Mws) {
    __shared__ float Wl[1024];
    __shared__ float S[2][CHUNKS][128];   // per-chunk signature state, double buffered
    __shared__ float R[2][128];           // running merge result

    const int tid = threadIdx.x;
    Wl[tid] = W[tid];
    __syncthreads();

    // Phase A: 16 chunks scanned in lockstep; thread -> (chunk, 2 elements).
    const int chunk = tid >> 6;       // 0..15
    const int lane  = tid & 63;
    const int e0 = lane, e1 = lane + 64;
    const bool v0 = (e0 < SIG_DIM), v1 = (e1 < SIG_DIM);
    int k0 = 1, j0 = 0, k1 = 1, j1 = 0;
    if (v0) decode_elem(e0, k0, j0);
    if (v1) decode_elem(e1, k1, j1);

    int cur = 0;
    for (int s = 0; s < CHUNK_LEN; ++s) {
        const int g = chunk * CHUNK_LEN + s;     // global increment index
        const float* Ain = &S[cur][chunk][0];
        float* Aout = &S[cur ^ 1][chunk][0];
        if (g < NINC) {
            const float dx0 = Wl[g + 1];         // channel 0 increment
            const float dx1 = Wl[513 + g];       // channel 1 increment
            const bool first = (s == 0);
            if (v0) Aout[e0] = chen_step_inc(Ain, e0, k0, j0, dx0, dx1, first);
            if (v1) Aout[e1] = chen_step_inc(Ain, e1, k1, j1, dx0, dx1, first);
        } else {                                  // tail of last chunk: carry forward
            if (v0) Aout[e0] = Ain[e0];
            if (v1) Aout[e1] = Ain[e1];
        }
        __syncthreads();
        cur ^= 1;
    }
    // CHUNK_LEN even -> final chunk states live in S[0][c][*].

    // Phase B: sequential Chen merge of the 16 chunk signatures (order matters).
    int kk = 1, jj = 0;
    if (tid < SIG_DIM) decode_elem(tid, kk, jj);
    if (tid < 128) R[0][tid] = (tid < SIG_DIM) ? S[0][0][tid] : 0.0f;
    __syncthreads();

    int rc = 0;
    for (int c = 1; c < CHUNKS; ++c) {
        float val = 0.0f;
        if (tid < SIG_DIM)
            val = chen_combine_sig(&R[rc][0], &S[0][c][0], tid, kk, jj);
        if (tid < 128) R[rc ^ 1][tid] = val;
        __syncthreads();
        rc ^= 1;
    }

    // Emit M[8][128] to workspace: row k holds sig entries of level k.
    if (tid < 128) {
        #pragma unroll
        for (int r = 0; r < 8; ++r) Mws[r * 128 + tid] = 0.0f;
        if (tid < SIG_DIM) Mws[kk * 128 + tid] = R[rc][tid];
    }
}

// out = V(4096x8) @ M(8x128), V[b,k] = x[b]^k.
// One wave per 16x16 output tile; two chained K=4 f32 WMMAs (K powers 0..3, 4..7).
__global__ __launch_bounds__(256)
void scale_wmma_kernel(const float* __restrict__ x, const float* __restrict__ M,
                       float* __restrict__ out) {
    const int tid  = threadIdx.x;
    const int wave = tid >> 5;            // N-tile: 0..7 (cols 16*wave..)
    const int lane = tid & 31;
    const int h    = lane >> 4;           // lane half
    const int l16  = lane & 15;
    const int row0 = blockIdx.x * 16;     // batch tile base (256 blocks * 16 = 4096)
    const int col  = wave * 16 + l16;     // 0..127 (M is padded to 128 cols)

    // A-matrix (16x4 f32): lane half h, VGPR r -> K = r + 2h; M-row = l16.
    const float xv = x[row0 + l16];
    const float x2 = xv * xv;
    const float x4 = x2 * x2;
    const float base = h ? x2 : 1.0f;
    v2f a0 = { base,      base * xv };        // K = {2h, 2h+1}
    v2f a1 = { base * x4, base * x4 * xv };   // K = {4+2h, 5+2h}

    // B-matrix (4x16 f32): VGPR r, half h -> K = r + 2h; N = col.
    v2f b0 = { M[(2 * h)     * 128 + col], M[(2 * h + 1) * 128 + col] };
    v2f b1 = { M[(4 + 2 * h) * 128 + col], M[(5 + 2 * h) * 128 + col] };

    v8f c = {};
    c = __builtin_amdgcn_wmma_f32_16x16x4_f32(false, a0, false, b0, (short)0, c, false, false);
    c = __builtin_amdgcn_wmma_f32_16x16x4_f32(false, a1, false, b1, (short)0, c, false, false);

    // C/D layout: VGPR r, half h -> row = r + 8h; col = l16 (+ tile offsets).
    if (col < SIG_DIM) {
        #pragma unroll
        for (int r = 0; r < 8; ++r) {
            const int row = row0 + r + 8 * h;
            out[row * SIG_DIM + col] = c[r];
        }
    }
}

extern "C" void kernel_launch(void* const* d_in, const int* in_sizes, int n_in,
                              void* d_out, int out_size, void* d_ws, size_t ws_size,
                              hipStream_t stream) {
    const float* x = (const float*)d_in[0];   // (4096, 1) fp32
    const float* W = (const float*)d_in[1];   // (1024, 1) fp32
    float* out = (float*)d_out;               // (4096, 126) fp32
    float* Mws = (float*)d_ws;                // 8*128 fp32 scratch

    sig_shared_kernel<<<1, 1024, 0, stream>>>(W, Mws);
    scale_wmma_kernel<<<256, 256, 0, stream>>>(x, Mws, out);
}